// GConv_38671885533578
// MI455X (gfx1250) — compile-verified
//
#include <hip/hip_runtime.h>
#include <hip/hip_bf16.h>

// Problem constants (match reference)
#define Nn 1024
#define Bb 4
#define Tt 12
#define Cc 64
#define Dd 64
#define Hh 2
#define Kk 3
#define Ee 16384
#define ETOT (Ee + Nn)          // 17408 edges per kernel (with self loops)
#define BT  (Bb * Tt)           // 48
#define Mm  (Nn * BT)           // 49152 rows in [N,B,T] flattening
#define HD  (Hh * Dd)           // 128
#define SLOPE_ATTN 0.2f
#define SLOPE_ACT  0.01f

typedef __attribute__((ext_vector_type(2))) float v2f;
typedef __attribute__((ext_vector_type(8))) float v8f;

// ---------------------------------------------------------------- utilities

__global__ void zero_kernel(float* __restrict__ p, int n) {
  int i = blockIdx.x * 256 + threadIdx.x;
  if (i < n) p[i] = 0.f;
}

// xT[m, c] = input[b, n, t, c] with m = (n*B + b)*T + t
__global__ void transpose_kernel(const float* __restrict__ in, float* __restrict__ xT) {
  int i = blockIdx.x * 256 + threadIdx.x;
  if (i >= Mm * Cc) return;
  int m = i >> 6, c = i & 63;
  int t = m % Tt, b = (m / Tt) % Bb, n = m / BT;
  xT[i] = in[((((size_t)b * Nn) + n) * Tt + t) * Cc + c];
}

// per-GAT-kernel scratch init: rst_un = 0, emax = -inf, denom = 0
__global__ void initk_kernel(float* __restrict__ rst, float* __restrict__ emax,
                             float* __restrict__ denom) {
  int i = blockIdx.x * 256 + threadIdx.x;
  if (i < Mm * HD) rst[i] = 0.f;
  if (i < Mm * Hh) { emax[i] = -__builtin_huge_valf(); denom[i] = 0.f; }
}

// -------------------------------------------------- fc GEMM: feat = xT @ fc_w[k]
// M=49152, K=64, Ncols=128. Block covers 32 rows x 128 cols; 4 waves,
// each wave: 2 col tiles x 2 row tiles (B fragments reused across row tiles).
#define ASTRIDE 68   // 64 + 4 pad: 16B-aligned rows, stride%64==4 -> bank-spread
__global__ __launch_bounds__(128) void fc_gemm_kernel(const float* __restrict__ xT,
                                                      const float* __restrict__ fcw,
                                                      float* __restrict__ feat) {
  __shared__ float As[32 * ASTRIDE];
  int row0 = blockIdx.x * 32;
  {
    const float4* src = (const float4*)(xT + (size_t)row0 * Cc);
    for (int i = threadIdx.x; i < 32 * 16; i += 128) {
      int r = i >> 4, c4 = i & 15;
      *(float4*)&As[r * ASTRIDE + c4 * 4] = src[r * 16 + c4];
    }
  }
  __syncthreads();
  int wave = threadIdx.x >> 5, lane = threadIdx.x & 31;
  int rowA = lane & 15;
  int kg = (lane >> 4) << 1;                 // lanes 0-15: K{0,1}; 16-31: K{2,3}
  for (int ct = wave; ct < 8; ct += 4) {     // uniform per wave: 2 iterations
    int colN = ct * 16 + (lane & 15);
    v8f acc0 = {}, acc1 = {};
#pragma unroll
    for (int s = 0; s < 16; ++s) {           // K = 64 in steps of 4
      int k0 = s * 4;
      v2f b;
      b.x = fcw[(k0 + kg) * HD + colN];
      b.y = fcw[(k0 + kg + 1) * HD + colN];
      v2f a0 = *(const v2f*)&As[rowA * ASTRIDE + k0 + kg];          // ds_load_b64
      v2f a1 = *(const v2f*)&As[(rowA + 16) * ASTRIDE + k0 + kg];
      acc0 = __builtin_amdgcn_wmma_f32_16x16x4_f32(false, a0, false, b,
                                                   (short)0, acc0, false, false);
      acc1 = __builtin_amdgcn_wmma_f32_16x16x4_f32(false, a1, false, b,
                                                   (short)0, acc1, false, false);
    }
    int rbase0 = row0 + ((lane >> 4) << 3);  // D rows: r + 8*(lane>=16)
#pragma unroll
    for (int r = 0; r < 8; ++r) {
      feat[(size_t)(rbase0 + r) * HD + colN]      = acc0[r];
      feat[(size_t)(rbase0 + 16 + r) * HD + colN] = acc1[r];
    }
  }
}

// -------------------------------------------- el/er = feat · attn_{l,r} over D
__global__ void attn_dot_kernel(const float* __restrict__ feat,
                                const float* __restrict__ al,
                                const float* __restrict__ ar,
                                float* __restrict__ el, float* __restrict__ er) {
  int i = blockIdx.x * 256 + threadIdx.x;
  if (i >= Mm * Hh) return;
  int m = i >> 1, h = i & 1;
  const float* f  = feat + (size_t)m * HD + h * Dd;
  const float* wl = al + h * Dd;
  const float* wr = ar + h * Dd;
  float sl = 0.f, sr = 0.f;
  for (int d = 0; d < Dd; ++d) { float fv = f[d]; sl += fv * wl[d]; sr += fv * wr[d]; }
  el[i] = sl; er[i] = sr;
}

// ----------------------------------- edge pass A: segment max of leaky(el+er)
__device__ __forceinline__ void atomicMaxF(float* addr, float v) {
  if (v >= 0.f) atomicMax((int*)addr, __float_as_int(v));
  else          atomicMin((unsigned int*)addr, __float_as_uint(v));
}

__global__ __launch_bounds__(128) void edge_max_kernel(const float* __restrict__ el,
                                                       const float* __restrict__ er,
                                                       const int* __restrict__ src,
                                                       const int* __restrict__ dst,
                                                       float* __restrict__ emax) {
  int tid = threadIdx.x;
  if (tid >= BT * Hh) return;
  int e = blockIdx.x;
  int s = src[e], dd = dst[e];
  int bt = tid >> 1, h = tid & 1;
  float ev = el[(s * BT + bt) * Hh + h] + er[(dd * BT + bt) * Hh + h];
  ev = ev > 0.f ? ev : SLOPE_ATTN * ev;
  atomicMaxF(&emax[(dd * BT + bt) * Hh + h], ev);
}

// ------------- edge pass B: a = exp(e-emax); denom += a; rst += a * feat[src]
__global__ __launch_bounds__(256) void edge_agg_kernel(const float* __restrict__ el,
                                                       const float* __restrict__ er,
                                                       const float* __restrict__ feat,
                                                       const int* __restrict__ src,
                                                       const int* __restrict__ dst,
                                                       const float* __restrict__ emax,
                                                       float* __restrict__ denom,
                                                       float* __restrict__ rst) {
  __shared__ float a_sh[BT * Hh];            // 96 softmax numerators
  int e = blockIdx.x;
  int tid = threadIdx.x;
  int s = src[e], dd = dst[e];
  if (tid < BT * Hh) {
    int bt = tid >> 1, h = tid & 1;
    int id = (dd * BT + bt) * Hh + h;
    float ev = el[(s * BT + bt) * Hh + h] + er[id];
    ev = ev > 0.f ? ev : SLOPE_ATTN * ev;
    float a = __expf(ev - emax[id]);
    a_sh[tid] = a;
    atomicAdd(&denom[id], a);
  }
  __syncthreads();
  for (int idx = tid; idx < BT * HD; idx += 256) {  // 6144 MACs per edge
    int bth = idx >> 6;                             // (bt*2 + h)
    int d = idx & 63;
    int bt = bth >> 1, h = bth & 1;
    float v = a_sh[bth] * feat[((size_t)s * BT + bt) * HD + h * Dd + d];
    atomicAdd(&rst[((size_t)dd * BT + bt) * HD + h * Dd + d], v);
  }
}

// ------------- finalize: result += weight[k] * (rst_un/denom + gat_bias[k])
__global__ void finalize_kernel(const float* __restrict__ rst,
                                const float* __restrict__ denom,
                                const float* __restrict__ bias,
                                const float* __restrict__ weight, int k,
                                float* __restrict__ result) {
  int i = blockIdx.x * 256 + threadIdx.x;
  if (i >= Mm * HD) return;
  int col = i & 127;
  int m = i >> 7;
  int h = col >> 6;
  float v = rst[i] / denom[m * Hh + h] + bias[col];
  result[i] += weight[k] * v;
}

// -------- merge GEMM + bias + leaky + residual + transpose-out: [B,N,T,D]
// M=49152, K=128, Ncols=64. Block covers 32 rows; 4 waves, each wave:
// 1 col tile x 2 row tiles (B fragments reused across row tiles).
#define ASTRIDE2 132  // 128 + 4 pad
__global__ __launch_bounds__(128) void merge_kernel(const float* __restrict__ result,
                                                    const float* __restrict__ mw,
                                                    const float* __restrict__ mb,
                                                    const float* __restrict__ xT,
                                                    float* __restrict__ out) {
  __shared__ float As[32 * ASTRIDE2];
  int row0 = blockIdx.x * 32;
  {
    const float4* src = (const float4*)(result + (size_t)row0 * HD);
    for (int i = threadIdx.x; i < 32 * 32; i += 128) {
      int r = i >> 5, c4 = i & 31;
      *(float4*)&As[r * ASTRIDE2 + c4 * 4] = src[r * 32 + c4];
    }
  }
  __syncthreads();
  int wave = threadIdx.x >> 5, lane = threadIdx.x & 31;
  int rowA = lane & 15;
  int kg = (lane >> 4) << 1;
  int colN = wave * 16 + (lane & 15);
  v8f acc0 = {}, acc1 = {};
#pragma unroll
  for (int s = 0; s < 32; ++s) {             // K = 128 in steps of 4
    int k0 = s * 4;
    v2f b;
    b.x = mw[(k0 + kg) * Dd + colN];
    b.y = mw[(k0 + kg + 1) * Dd + colN];
    v2f a0 = *(const v2f*)&As[rowA * ASTRIDE2 + k0 + kg];
    v2f a1 = *(const v2f*)&As[(rowA + 16) * ASTRIDE2 + k0 + kg];
    acc0 = __builtin_amdgcn_wmma_f32_16x16x4_f32(false, a0, false, b,
                                                 (short)0, acc0, false, false);
    acc1 = __builtin_amdgcn_wmma_f32_16x16x4_f32(false, a1, false, b,
                                                 (short)0, acc1, false, false);
  }
  float bias = mb[colN];
  int rbase = (lane >> 4) << 3;
#pragma unroll
  for (int rt = 0; rt < 2; ++rt) {
#pragma unroll
    for (int r = 0; r < 8; ++r) {
      int m = row0 + rt * 16 + rbase + r;
      float v = (rt == 0 ? acc0[r] : acc1[r]) + bias;
      v = v > 0.f ? v : SLOPE_ACT * v;
      v += xT[(size_t)m * Cc + colN];        // residual (C == D)
      int t = m % Tt, b2 = (m / Tt) % Bb, n = m / BT;
      out[((((size_t)b2 * Nn) + n) * Tt + t) * Dd + colN] = v;
    }
  }
}

// ---------------------------------------------------------------- launcher

extern "C" void kernel_launch(void* const* d_in, const int* in_sizes, int n_in,
                              void* d_out, int out_size, void* d_ws, size_t ws_size,
                              hipStream_t stream) {
  const float* x   = (const float*)d_in[0];  // [B,N,T,C]
  const float* fcw = (const float*)d_in[1];  // [K,C,H*D]
  const float* al  = (const float*)d_in[2];  // [K,H,D]
  const float* ar  = (const float*)d_in[3];  // [K,H,D]
  const float* gb  = (const float*)d_in[4];  // [K,H*D]
  const float* wgt = (const float*)d_in[5];  // [K]
  const float* mw  = (const float*)d_in[6];  // [H*D,D]
  const float* mb  = (const float*)d_in[7];  // [D]
  const int* srcI  = (const int*)d_in[8];    // [K,ETOT]
  const int* dstI  = (const int*)d_in[9];    // [K,ETOT]
  float* out = (float*)d_out;                // [B,N,T,D]

  float* ws     = (float*)d_ws;
  float* xT     = ws;                            // M*64
  float* feat   = xT     + (size_t)Mm * Cc;      // M*128
  float* rst    = feat   + (size_t)Mm * HD;      // M*128
  float* result = rst    + (size_t)Mm * HD;      // M*128
  float* el     = result + (size_t)Mm * HD;      // M*H
  float* er     = el     + (size_t)Mm * Hh;
  float* emax   = er     + (size_t)Mm * Hh;
  float* denom  = emax   + (size_t)Mm * Hh;

  const int big = (Mm * HD + 255) / 256;         // 24576 blocks

  transpose_kernel<<<(Mm * Cc + 255) / 256, 256, 0, stream>>>(x, xT);
  zero_kernel<<<big, 256, 0, stream>>>(result, Mm * HD);

  for (int k = 0; k < Kk; ++k) {
    initk_kernel<<<big, 256, 0, stream>>>(rst, emax, denom);
    fc_gemm_kernel<<<Mm / 32, 128, 0, stream>>>(xT, fcw + (size_t)k * Cc * HD, feat);
    attn_dot_kernel<<<(Mm * Hh + 255) / 256, 256, 0, stream>>>(
        feat, al + k * Hh * Dd, ar + k * Hh * Dd, el, er);
    edge_max_kernel<<<ETOT, 128, 0, stream>>>(
        el, er, srcI + (size_t)k * ETOT, dstI + (size_t)k * ETOT, emax);
    edge_agg_kernel<<<ETOT, 256, 0, stream>>>(
        el, er, feat, srcI + (size_t)k * ETOT, dstI + (size_t)k * ETOT,
        emax, denom, rst);
    finalize_kernel<<<big, 256, 0, stream>>>(rst, denom, gb + k * HD, wgt, k, result);
  }

  merge_kernel<<<Mm / 32, 128, 0, stream>>>(result, mw, mb, xT, out);
}